// LoraEmbedding_31499290149560
// MI455X (gfx1250) — compile-verified
//
#include <hip/hip_runtime.h>
#include <hip/hip_bf16.h>

// LoRA embedding for MI455X (gfx1250, wave32):
//   out[t,:] = E[idx[t],:] + (E[idx[t],:] @ A) @ B + bias
// Bandwidth-bound (AI ~4 FLOP/B; floor ~12us at 23.3 TB/s).
// Matrix work via V_WMMA_F32_16X16X4_F32; A-weights staged in LDS so the
// stage-1 stream loop is branch-free: 1x global_load_b64 + 2x ds_load_b32
// + 2x v_cndmask + 1x wmma per k-step.

typedef __attribute__((ext_vector_type(2))) float v2f;
typedef __attribute__((ext_vector_type(8))) float v8f;

#define FDIM 1024
#define RANK 8
#define WAVES_PER_BLOCK 4
#define TOKENS_PER_WAVE 16
#define TOKENS_PER_BLOCK (WAVES_PER_BLOCK * TOKENS_PER_WAVE)

__global__ __launch_bounds__(WAVES_PER_BLOCK * 32)
void lora_embedding_kernel(const int* __restrict__ idx,
                           const float* __restrict__ emb,
                           const float* __restrict__ Alora,   // [FDIM, RANK]
                           const float* __restrict__ Blora,   // [RANK, FDIM]
                           const float* __restrict__ bias,    // [FDIM]
                           float* __restrict__ out,           // [ntok, FDIM]
                           int ntok)
{
    __shared__ float sA[FDIM * RANK];                  // 32 KB, k-major copy of A
    __shared__ float lds_low[WAVES_PER_BLOCK][TOKENS_PER_WAVE * 16];  // 4 KB

    // Cooperative A copy (32 KB, one-time; coalesced b128 both sides).
    {
        const float4* __restrict__ src = (const float4*)Alora;
        float4* __restrict__ dst = (float4*)sA;
        for (int i = threadIdx.x; i < (FDIM * RANK) / 4; i += WAVES_PER_BLOCK * 32)
            dst[i] = src[i];
    }
    __syncthreads();

    const int lane = threadIdx.x & 31;
    const int wave = threadIdx.x >> 5;
    const int tok0 = (blockIdx.x * WAVES_PER_BLOCK + wave) * TOKENS_PER_WAVE;
    if (tok0 >= ntok) return;   // whole-wave exit; EXEC all-ones for WMMA

    const int m  = lane & 15;   // A-frag row (token in tile) / B-frag column
    const int hi = lane >> 4;   // 0: lanes 0-15, 1: lanes 16-31
    const int mr = m & 7;       // clamped rank index (always in-bounds)
    const bool mv = (m < RANK); // zero-pad predicate for rank 8 -> 16

    // Row this lane streams for the stage-1 A-fragment.
    const long long rowoff = (long long)idx[tok0 + m] * FDIM;
    const float* __restrict__ rowp = emb + rowoff;

    // ---------------- Stage 1: low[16x8] = base[16x1024] @ A[1024x8] ----------------
    // A-frag (f32 A 16x4, ISA 7.12.2): lane<16 -> K=k0,k0+1 ; lane>=16 -> K=k0+2,k0+3
    // B-frag (4x16): lanes 0-15 rows k0,k0+1 ; lanes 16-31 rows k0+2,k0+3.
    v8f acc = {};
#pragma unroll 8
    for (int k0 = 0; k0 < FDIM; k0 += 4) {
        const int koff = k0 + (hi << 1);
        v2f a;
        a.x = rowp[koff];
        a.y = rowp[koff + 1];
        // Unconditional LDS loads + register select (no EXEC branching).
        const float w0 = sA[koff * RANK + mr];
        const float w1 = sA[(koff + 1) * RANK + mr];
        v2f b;
        b.x = mv ? w0 : 0.0f;
        b.y = mv ? w1 : 0.0f;
        acc = __builtin_amdgcn_wmma_f32_16x16x4_f32(false, a, false, b,
                                                    (short)0, acc, false, false);
    }

    // D-layout: lane L, vgpr p -> low[M = p + 8*hi][n = L&15]. Park in LDS,
    // reread in A-fragment layout (DS ops in-order within a wave).
    float* __restrict__ lw = lds_low[wave];
#pragma unroll
    for (int p = 0; p < 8; ++p)
        lw[(p + 8 * hi) * 16 + m] = acc[p];

    const int ka = hi << 1;
    v2f la0, la1;
    la0.x = lw[m * 16 + ka];
    la0.y = lw[m * 16 + ka + 1];
    la1.x = lw[m * 16 + 4 + ka];
    la1.y = lw[m * 16 + 5 + ka];

    // Hoist the 8 gathered row offsets this lane needs for the C fragment.
    long long rowc[8];
#pragma unroll
    for (int p = 0; p < 8; ++p)
        rowc[p] = (long long)idx[tok0 + p + 8 * hi] * FDIM;

    // ---------------- Stage 2: out = base + low @ B + bias ----------------
    for (int j = 0; j < FDIM; j += 16) {
        const int col = j + m;
        const float bb = bias[col];

        v2f b0, b1;                          // B rows ka..ka+1 and 4+ka..5+ka (all valid)
        b0.x = Blora[(ka)     * FDIM + col];
        b0.y = Blora[(ka + 1) * FDIM + col];
        b1.x = Blora[(4 + ka) * FDIM + col];
        b1.y = Blora[(5 + ka) * FDIM + col];

        v8f c;                               // C = base tile + bias (cache-resident reread)
#pragma unroll
        for (int p = 0; p < 8; ++p)
            c[p] = emb[rowc[p] + col] + bb;

        v8f d = __builtin_amdgcn_wmma_f32_16x16x4_f32(false, la0, false, b0,
                                                      (short)0, c, false, false);
        d = __builtin_amdgcn_wmma_f32_16x16x4_f32(false, la1, false, b1,
                                                  (short)0, d, false, false);

#pragma unroll
        for (int p = 0; p < 8; ++p)
            out[(long long)(tok0 + p + 8 * hi) * FDIM + col] = d[p];
    }
}

extern "C" void kernel_launch(void* const* d_in, const int* in_sizes, int n_in,
                              void* d_out, int out_size, void* d_ws, size_t ws_size,
                              hipStream_t stream) {
    const int*   idx   = (const int*)  d_in[0];   // [8, 4096] token ids
    const float* emb   = (const float*)d_in[1];   // [50257, 1024]
    const float* Al    = (const float*)d_in[2];   // [1024, 8]
    const float* Bl    = (const float*)d_in[3];   // [8, 1024]
    const float* bias  = (const float*)d_in[4];   // [1024]
    float*       out   = (float*)d_out;           // [8*4096, 1024]

    const int ntok = in_sizes[0];                 // 32768
    const int blocks = (ntok + TOKENS_PER_BLOCK - 1) / TOKENS_PER_BLOCK;

    lora_embedding_kernel<<<blocks, WAVES_PER_BLOCK * 32, 0, stream>>>(
        idx, emb, Al, Bl, bias, out, ntok);
}